// SpatialGCN_88227218195280
// MI455X (gfx1250) — compile-verified
//
#include <hip/hip_runtime.h>
#include <hip/hip_bf16.h>

typedef __attribute__((ext_vector_type(2))) float v2f;
typedef __attribute__((ext_vector_type(4))) float v4f;
typedef __attribute__((ext_vector_type(8))) float v8f;

static constexpr float C_SIGMA = 5.0e-7f;   // 1/(2*1000^2)
static constexpr int SPLITK = 4;            // K-split for wave parallelism

// ---------------- small prep kernels ----------------

__global__ void k_zero(float* __restrict__ p, int n) {
  int i = blockIdx.x * blockDim.x + threadIdx.x;
  if (i < n) p[i] = 0.0f;
}

__global__ void k_row_norms(const float* __restrict__ pos, float* __restrict__ r, int n) {
  int i = blockIdx.x * blockDim.x + threadIdx.x;
  if (i < n) {
    float x = pos[2 * i], y = pos[2 * i + 1];
    r[i] = x * x + y * y;
  }
}

__global__ void k_deg_init(float* __restrict__ deg, int n) {
  int i = blockIdx.x * blockDim.x + threadIdx.x;
  if (i < n) deg[i] = 1.0f;   // self loop
}

__global__ void k_deg_accum(const int* __restrict__ dst, float* __restrict__ deg, int e) {
  int i = blockIdx.x * blockDim.x + threadIdx.x;
  if (i < e) unsafeAtomicAdd(&deg[dst[i]], 1.0f);
}

__global__ void k_dinv(float* __restrict__ deg, int n) {
  int i = blockIdx.x * blockDim.x + threadIdx.x;
  if (i < n) deg[i] = rsqrtf(deg[i]);   // deg >= 1 always (self loops)
}

// ---------------- fused Gaussian-kernel x dense-matmul (WMMA f32) ----------------
// out[N x C] += exp(-max(r_i + r_j - 2*pos_i.pos_j, 0)*c)[K-chunk] @ hin[K-chunk x C]
// One wave computes a 32-row strip (2 row tiles) across all C columns:
//   RT*CT wmma accumulators; B fragments shared across the 2 row tiles.
// Grid.y = split-K chunk; partials accumulated with f32 HW atomics.
// A-fragment layout (ISA 7.12.2, 32-bit A 16x4): lane half h holds K = 2h, 2h+1.
// B-fragment: VGPR v holds K = v + 2h, N = lane&15 (mirror of C-matrix layout).

template <int C>
__global__ __launch_bounds__(128) void k_kernel_matmul(
    const float* __restrict__ pos, const float* __restrict__ rbuf,
    const float* __restrict__ hin, float* __restrict__ out, int n, int kChunk) {
  constexpr int CT = C / 16;
  constexpr int RT = 2;                      // row tiles per wave
  const int lane = threadIdx.x & 31;
  const int wave = threadIdx.x >> 5;
  const int strip = blockIdx.x * 4 + wave;   // 32-row strip
  const int i0 = strip * 16 * RT;
  if (i0 >= n) return;

  const int m = lane & 15;       // row within A-tile / col within B-tile
  const int h = lane >> 4;       // wave half

  float xi[RT], yi[RT], ri[RT];
#pragma unroll
  for (int r = 0; r < RT; ++r) {
    int i = i0 + r * 16 + m;
    xi[r] = pos[2 * i];
    yi[r] = pos[2 * i + 1];
    ri[r] = rbuf[i];
  }

  v8f acc[RT][CT];
#pragma unroll
  for (int r = 0; r < RT; ++r)
#pragma unroll
    for (int t = 0; t < CT; ++t) acc[r][t] = (v8f)(0.0f);

  const int k0 = blockIdx.y * kChunk;
  const int k1 = k0 + kChunk;
  for (int kb = k0; kb < k1; kb += 4) {
    const int j = kb + 2 * h;                         // this lane covers K = j, j+1
    const v4f pj = *(const v4f*)(pos + 2 * j);        // pos[j].xy, pos[j+1].xy
    const v2f rj = *(const v2f*)(rbuf + j);

    v2f a[RT];
#pragma unroll
    for (int r = 0; r < RT; ++r) {
      float sq0 = fmaxf(ri[r] + rj.x - 2.0f * (xi[r] * pj.x + yi[r] * pj.y), 0.0f);
      float sq1 = fmaxf(ri[r] + rj.y - 2.0f * (xi[r] * pj.z + yi[r] * pj.w), 0.0f);
      a[r].x = __expf(-sq0 * C_SIGMA);
      a[r].y = __expf(-sq1 * C_SIGMA);
    }

    const float* __restrict__ brow = hin + (size_t)j * C + m;
    v2f b[CT];
#pragma unroll
    for (int t = 0; t < CT; ++t) {
      b[t].x = brow[t * 16];        // hin[j,   t*16+m]
      b[t].y = brow[C + t * 16];    // hin[j+1, t*16+m]
    }

#pragma unroll
    for (int r = 0; r < RT; ++r)
#pragma unroll
      for (int t = 0; t < CT; ++t)
        acc[r][t] = __builtin_amdgcn_wmma_f32_16x16x4_f32(
            /*neg_a=*/false, a[r], /*neg_b=*/false, b[t],
            /*c_mod=*/(short)0, acc[r][t], /*reuse_a=*/false, /*reuse_b=*/false);
  }

  // C-matrix layout: lane L, VGPR v -> row = v + 8*(L>>4), col = L&15
#pragma unroll
  for (int r = 0; r < RT; ++r)
#pragma unroll
    for (int t = 0; t < CT; ++t)
#pragma unroll
      for (int v = 0; v < 8; ++v) {
        int row = i0 + r * 16 + h * 8 + v;
        unsafeAtomicAdd(&out[(size_t)row * C + t * 16 + m], acc[r][t][v]);
      }
}

// ---------------- linear (kx @ W) + bias + self-loop init ----------------

template <int CIN, int COUT>
__global__ void k_linear_init(const float* __restrict__ kx, const float* __restrict__ W,
                              const float* __restrict__ b, const float* __restrict__ dinv,
                              float* __restrict__ xw, float* __restrict__ outInit, int n) {
  int idx = blockIdx.x * blockDim.x + threadIdx.x;
  if (idx >= n * COUT) return;
  int node = idx / COUT, co = idx % COUT;
  const float* __restrict__ row = kx + (size_t)node * CIN;
  float s = 0.0f;
#pragma unroll
  for (int ci = 0; ci < CIN; ++ci) s = fmaf(row[ci], W[ci * COUT + co], s);
  xw[idx] = s;
  float d = dinv[node];
  outInit[idx] = d * d * s + b[co];   // self-loop contribution + bias
}

// ---------------- edge scatter: out[dst] += dinv[src]*dinv[dst]*xw[src] ----------------

template <int COUT>
__global__ void k_scatter(const int* __restrict__ src, const int* __restrict__ dst,
                          const float* __restrict__ dinv, const float* __restrict__ xw,
                          float* __restrict__ out, int e) {
  int eidx = blockIdx.x * blockDim.x + threadIdx.x;
  if (eidx >= e) return;
  int s = src[eidx], d = dst[eidx];
  float w = dinv[s] * dinv[d];
  const float* __restrict__ xs = xw + (size_t)s * COUT;
  float* __restrict__ od = out + (size_t)d * COUT;
#pragma unroll
  for (int c = 0; c < COUT; ++c) unsafeAtomicAdd(od + c, w * xs[c]);
}

__global__ void k_relu(float* __restrict__ a, int n) {
  int i = blockIdx.x * blockDim.x + threadIdx.x;
  if (i < n) a[i] = fmaxf(a[i], 0.0f);
}

__global__ void k_logsoftmax16(const float* __restrict__ in, float* __restrict__ out, int n) {
  int i = blockIdx.x * blockDim.x + threadIdx.x;
  if (i >= n) return;
  const float* __restrict__ r = in + (size_t)i * 16;
  float mx = r[0];
#pragma unroll
  for (int c = 1; c < 16; ++c) mx = fmaxf(mx, r[c]);
  float s = 0.0f;
#pragma unroll
  for (int c = 0; c < 16; ++c) s += __expf(r[c] - mx);
  float lse = mx + __logf(s);
#pragma unroll
  for (int c = 0; c < 16; ++c) out[(size_t)i * 16 + c] = r[c] - lse;
}

// ---------------- driver ----------------

extern "C" void kernel_launch(void* const* d_in, const int* in_sizes, int n_in,
                              void* d_out, int out_size, void* d_ws, size_t ws_size,
                              hipStream_t stream) {
  const float* x   = (const float*)d_in[0];   // [N,64]
  const float* pos = (const float*)d_in[1];   // [N,2]
  const int*   ei  = (const int*)d_in[2];     // [2,E]
  const float* W1  = (const float*)d_in[3];
  const float* b1  = (const float*)d_in[4];
  const float* W2  = (const float*)d_in[5];
  const float* b2  = (const float*)d_in[6];
  const float* W3  = (const float*)d_in[7];
  const float* b3  = (const float*)d_in[8];
  float* out = (float*)d_out;

  const int n = in_sizes[0] / 64;   // 8192
  const int e = in_sizes[2] / 2;    // 262144
  const int* esrc = ei;
  const int* edst = ei + e;

  float* deg   = (float*)d_ws;          // [n] degree -> dinv in place
  float* rbuf  = deg + n;               // [n] pos row norms
  float* bufKX = rbuf + n;              // [n,64] kernel @ h
  float* bufXW = bufKX + (size_t)n*64;  // [n,32] (kernel@h) @ W
  float* bufH  = bufXW + (size_t)n*32;  // [n,32] layer output

  const int T = 256;
  const int gN = (n + T - 1) / T;
  const int gE = (e + T - 1) / T;
  const int kChunk = n / SPLITK;
  // 4 waves per 128-thread block, each wave owns a 32-row strip; grid.y = K-split
  const dim3 gMM((n / 32 + 3) / 4, SPLITK);

  // normalization + pos norms
  k_row_norms<<<gN, T, 0, stream>>>(pos, rbuf, n);
  k_deg_init <<<gN, T, 0, stream>>>(deg, n);
  k_deg_accum<<<gE, T, 0, stream>>>(edst, deg, e);
  k_dinv     <<<gN, T, 0, stream>>>(deg, n);

  // ---- layer 1: h1 = relu(gcn(kernel @ x, W1, b1)) ----
  k_zero<<<(n * 64 + T - 1) / T, T, 0, stream>>>(bufKX, n * 64);
  k_kernel_matmul<64><<<gMM, 128, 0, stream>>>(pos, rbuf, x, bufKX, n, kChunk);
  k_linear_init<64, 32><<<(n * 32 + T - 1) / T, T, 0, stream>>>(bufKX, W1, b1, deg, bufXW, bufH, n);
  k_scatter<32><<<gE, T, 0, stream>>>(esrc, edst, deg, bufXW, bufH, e);
  k_relu<<<(n * 32 + T - 1) / T, T, 0, stream>>>(bufH, n * 32);

  // ---- layer 2: h2 = relu(gcn(kernel @ h1, W2, b2)) ----
  k_zero<<<(n * 32 + T - 1) / T, T, 0, stream>>>(bufKX, n * 32);
  k_kernel_matmul<32><<<gMM, 128, 0, stream>>>(pos, rbuf, bufH, bufKX, n, kChunk);
  k_linear_init<32, 32><<<(n * 32 + T - 1) / T, T, 0, stream>>>(bufKX, W2, b2, deg, bufXW, bufH, n);
  k_scatter<32><<<gE, T, 0, stream>>>(esrc, edst, deg, bufXW, bufH, e);
  k_relu<<<(n * 32 + T - 1) / T, T, 0, stream>>>(bufH, n * 32);

  // ---- layer 3: h3 = gcn(kernel @ h2, W3, b3); log_softmax ----
  k_zero<<<(n * 32 + T - 1) / T, T, 0, stream>>>(bufKX, n * 32);
  k_kernel_matmul<32><<<gMM, 128, 0, stream>>>(pos, rbuf, bufH, bufKX, n, kChunk);
  k_linear_init<32, 16><<<(n * 16 + T - 1) / T, T, 0, stream>>>(bufKX, W3, b3, deg, bufXW, bufH, n);
  k_scatter<16><<<gE, T, 0, stream>>>(esrc, edst, deg, bufXW, bufH, e);
  k_logsoftmax16<<<gN, T, 0, stream>>>(bufH, out, n);
}